// PointerNet_5463198400975
// MI455X (gfx1250) — compile-verified
//
#include <hip/hip_runtime.h>
#include <math.h>

// ---------- types ----------
typedef float v2f __attribute__((ext_vector_type(2)));
typedef float v8f __attribute__((ext_vector_type(8)));
typedef unsigned int v4u __attribute__((ext_vector_type(4)));
typedef int v4i __attribute__((ext_vector_type(4)));
typedef int v8i __attribute__((ext_vector_type(8)));

#if __has_builtin(__builtin_amdgcn_tanhf)
#define FAST_TANH(x) __builtin_amdgcn_tanhf(x)
#else
#define FAST_TANH(x) tanhf(x)
#endif

#if __has_builtin(__builtin_amdgcn_tensor_load_to_lds) && \
    __has_builtin(__builtin_amdgcn_s_wait_tensorcnt)
#define USE_TDM 1
#else
#define USE_TDM 0
#endif

#define HH 64          // hidden size H
#define TE 512         // encoder length
#define CTX_STRIDE 65  // padded LDS row stride (65 mod 64 = 1 -> conflict-free)

// =====================================================================
// Kernel A: out[M, 64] = A[M, K] @ W[K, 64] + bias, fp32 WMMA 16x16x4.
// One wave per 16x16 output tile; block = 128 threads = 4 waves = full
// N=64 for a 16-row strip. Grid = M/16.
// 32-bit A 16x4 layout:  lanes 0-15 hold K=k,k+1; lanes 16-31 hold K=k+2,k+3
// 32-bit B 4x16 layout:  VGPR v, lane l: K = k + (l/16)*2 + v, N = l%16
// 32-bit D 16x16 layout: VGPR v, lane l: M = (l/16)*8 + v,     N = l%16
// =====================================================================
__global__ __launch_bounds__(128) void proj_wmma_f32(
    const float* __restrict__ A, const float* __restrict__ W,
    const float* __restrict__ bias, float* __restrict__ out,
    int M, int K) {
  const int m0   = blockIdx.x * 16;
  const int wave = threadIdx.x >> 5;
  const int lane = threadIdx.x & 31;
  const int n0   = wave * 16;
  const int half = lane >> 4;   // 0: lanes 0-15, 1: lanes 16-31
  const int lm   = lane & 15;
  const int arow = m0 + lm;     // A row this lane feeds
  const int bn   = n0 + lm;     // output column this lane feeds

  const float* __restrict__ arp = A + (long)arow * K;
  v8f c = {};
  for (int k = 0; k < K; k += 4) {
    const int ka = k + half * 2;
    v2f a, b;
    a.x = arp[ka];
    a.y = arp[ka + 1];
    b.x = W[(long)(ka)     * HH + bn];
    b.y = W[(long)(ka + 1) * HH + bn];
    // D = A x B + C, fp32 in / fp32 accumulate
    c = __builtin_amdgcn_wmma_f32_16x16x4_f32(
        /*neg_a=*/false, a, /*neg_b=*/false, b,
        /*c_mod=*/(short)0, c, /*reuse_a=*/false, /*reuse_b=*/false);
  }
  const float bv = bias[bn];
#pragma unroll
  for (int v = 0; v < 8; ++v) {
    const int row = m0 + half * 8 + v;
    out[(long)row * HH + bn] = c[v] + bv;
  }
}

// =====================================================================
// Kernel B: fused additive-attention scores + softmax.
// Grid: B * (Td/8) workgroups, 256 threads (8 waves).
// ctx_t[b] (512x64 f32) is DMA'd into LDS by the Tensor Data Mover with
// a 1-DWORD pad per 64-DWORD row -> stride-65 conflict-free layout.
// Wave w owns row td = td0 + w; each lane owns 16 te positions.
// Softmax is done entirely with wave32 shuffles.
// =====================================================================
__global__ __launch_bounds__(256) void attn_score_softmax(
    const float* __restrict__ ctx_t,   // [B, Te, H]
    const float* __restrict__ dec_t,   // [B, Td, H]
    const float* __restrict__ w2,      // [H]
    float* __restrict__ out,           // [B, Td, Te]
    int Td) {
  __shared__ float s_ctx[TE * CTX_STRIDE];  // ~130 KB
  __shared__ float s_dec[8 * HH];
  __shared__ float s_w2[HH];

  const int b    = blockIdx.x >> 6;         // Td/8 = 64 blocks per batch
  const int td0  = (blockIdx.x & 63) * 8;
  const int tid  = threadIdx.x;
  const int w    = tid >> 5;
  const int lane = tid & 31;

#if USE_TDM
  // ---- stage ctx_t[b] via Tensor Data Mover (wave 0 only) ----
  if (tid < 32) {
    const unsigned lds_base = (unsigned)(uintptr_t)(void*)s_ctx;  // flat->LDS offset
    const unsigned long long ga =
        (unsigned long long)(uintptr_t)(ctx_t + (size_t)b * TE * HH);
    v4u g0;
    g0.x = 1u;                                   // count=1, user descriptor
    g0.y = lds_base;                             // lds_addr (bytes)
    g0.z = (unsigned)(ga & 0xffffffffu);         // global_addr[31:0]
    g0.w = (unsigned)((ga >> 32) & 0x1ffffffu)   // global_addr[56:32]
           | (2u << 30);                         // type = 2 ("image")
    v8i g1;
    g1[0] = (2 << 16)      // data_size = 2 -> 4 bytes
          | (1 << 20)      // pad_enable
          | (5 << 22);     // pad_interval = 5 -> every 64 DWORDs
                           // pad_amount  = 0 -> 1 DWORD  => LDS stride 65
    g1[1] = (HH << 16);    // atomic_barrier_addr=0 | tensor_dim0[15:0]=64
    g1[2] = (TE << 16);    // tensor_dim0[31:16]=0 | tensor_dim1[15:0]=512
    g1[3] = (HH << 16);    // tensor_dim1[31:16]=0 | tile_dim0=64
    g1[4] = TE;            // tile_dim1=512 | tile_dim2=0
    g1[5] = HH;            // tensor_dim0_stride[31:0]=64 (contiguous rows)
    g1[6] = 0;             // dim0_stride[47:32]=0 | dim1_stride[15:0]=0
    g1[7] = 0;
    v4i g2 = {0, 0, 0, 0};          // tile_dim3=0 -> unused (2D tensor)
    v4i g3 = {0, 0, 0, 0};          // tile_dim4=0 -> unused
    v8i g4 = {0, 0, 0, 0, 0, 0, 0, 0};  // unused trailing group (VADDR4/NULL)
    __builtin_amdgcn_tensor_load_to_lds(g0, g1, g2, g3, g4, /*cpol=*/0);
    __builtin_amdgcn_s_wait_tensorcnt(0);        // DMA done before barrier
  }
#else
  // ---- fallback: manual staging (vectorized global reads) ----
  {
    const float4* __restrict__ cb4 = (const float4*)(ctx_t + (long)b * TE * HH);
    for (int idx = tid; idx < TE * HH / 4; idx += 256) {
      const int te = idx >> 4;
      const int h4 = (idx & 15) * 4;
      const float4 v = cb4[idx];
      float* dst = &s_ctx[te * CTX_STRIDE + h4];
      dst[0] = v.x; dst[1] = v.y; dst[2] = v.z; dst[3] = v.w;
    }
  }
#endif

  if (tid < HH) s_w2[tid] = w2[tid];
  // each wave stages its own decoder row (64 floats)
  {
    const float* __restrict__ drow = dec_t + ((long)(b * Td + td0 + w)) * HH;
    s_dec[w * HH + lane]      = drow[lane];
    s_dec[w * HH + lane + 32] = drow[lane + 32];
  }
  __syncthreads();

  // ---- scores: acc[i] = sum_h w2[h]*tanh(dec[h] + ctx[te][h]) ----
  float acc[16];
#pragma unroll
  for (int i = 0; i < 16; ++i) acc[i] = 0.0f;

  const float* __restrict__ sd = s_dec + w * HH;
  for (int h = 0; h < HH; ++h) {
    const float dh = sd[h];     // broadcast LDS read
    const float wh = s_w2[h];   // broadcast LDS read
#pragma unroll
    for (int i = 0; i < 16; ++i) {
      const float x = dh + s_ctx[(i * 32 + lane) * CTX_STRIDE + h];
      acc[i] += wh * FAST_TANH(x);
    }
  }
  // (b2 is added uniformly to every score -> cancels in softmax)

  // ---- in-wave softmax over 512 values (16 per lane) ----
  float m = acc[0];
#pragma unroll
  for (int i = 1; i < 16; ++i) m = fmaxf(m, acc[i]);
#pragma unroll
  for (int off = 16; off >= 1; off >>= 1)
    m = fmaxf(m, __shfl_xor(m, off, 32));

  float e[16];
  float sum = 0.0f;
#pragma unroll
  for (int i = 0; i < 16; ++i) {
    e[i] = __expf(acc[i] - m);
    sum += e[i];
  }
#pragma unroll
  for (int off = 16; off >= 1; off >>= 1)
    sum += __shfl_xor(sum, off, 32);
  const float inv = 1.0f / sum;

  float* __restrict__ orow = out + ((long)(b * Td + td0 + w)) * TE;
#pragma unroll
  for (int i = 0; i < 16; ++i)
    orow[i * 32 + lane] = e[i] * inv;   // coalesced 128B stores per i
}

// =====================================================================
// launch
// =====================================================================
extern "C" void kernel_launch(void* const* d_in, const int* in_sizes, int n_in,
                              void* d_out, int out_size, void* d_ws, size_t ws_size,
                              hipStream_t stream) {
  (void)in_sizes; (void)n_in; (void)out_size; (void)ws_size;
  const float* ctx = (const float*)d_in[0];   // [4, 512, 256]
  const float* dec = (const float*)d_in[1];   // [4, 512, 512]
  const float* W1i = (const float*)d_in[2];   // [256, 64]
  const float* b1i = (const float*)d_in[3];   // [64]
  const float* W1h = (const float*)d_in[4];   // [512, 64]
  const float* b1h = (const float*)d_in[5];   // [64]
  const float* w2  = (const float*)d_in[6];   // [64]
  // d_in[7] = b2 scalar: uniform additive shift, cancels in softmax
  float* out = (float*)d_out;                 // [4, 512, 512]

  const int B = 4, Td = 512, E = 256, DE = 512;
  const int Mrows = B * Td;                   // 2048 rows for both GEMMs

  float* ctx_t = (float*)d_ws;                // [B*Te, 64] = 512 KB
  float* dec_t = ctx_t + (size_t)B * TE * HH; // [B*Td, 64] = 512 KB

  // WMMA projections: grid = M/16 strips, 4 waves cover N=64
  proj_wmma_f32<<<Mrows / 16, 128, 0, stream>>>(ctx, W1i, b1i, ctx_t, Mrows, E);
  proj_wmma_f32<<<Mrows / 16, 128, 0, stream>>>(dec, W1h, b1h, dec_t, Mrows, DE);

  // Fused scores + softmax: one WG per (b, 8 decoder rows)
  attn_score_softmax<<<B * (Td / 8), 256, 0, stream>>>(ctx_t, dec_t, w2, out, Td);
}